// EdgeLoss_17703855194348
// MI455X (gfx1250) — compile-verified
//
#include <hip/hip_runtime.h>
#include <hip/hip_bf16.h>
#include <math.h>

typedef __attribute__((ext_vector_type(2))) float v2f;
typedef __attribute__((ext_vector_type(8))) float v8f;

#define GS 512
#define GS2 (GS * GS)
#define SIGMA_F 2.0f

// ---------- sortable-uint encoding for float atomic min/max ----------
__device__ __forceinline__ unsigned encf(float f) {
    unsigned u = __float_as_uint(f);
    return (u & 0x80000000u) ? ~u : (u | 0x80000000u);
}
__device__ __forceinline__ float decf(unsigned e) {
    unsigned u = (e & 0x80000000u) ? (e & 0x7FFFFFFFu) : ~e;
    return __uint_as_float(u);
}

// ---------- per-batch bounding-box / grid params (recomputed cheaply) ----------
struct BP { float xmin, xmax, ymin, ymax; int gsx, gsy; };

__device__ __forceinline__ BP get_params(const unsigned* __restrict__ mm, int b) {
    BP p;
    float xmn = decf(mm[4 * b + 0]), xmx = decf(mm[4 * b + 1]);
    float ymn = decf(mm[4 * b + 2]), ymx = decf(mm[4 * b + 3]);
    if (xmx - xmn < 1e-6f) { xmn -= 0.5f; xmx += 0.5f; }
    if (ymx - ymn < 1e-6f) { ymn -= 0.5f; ymx += 0.5f; }
    float xr = xmx - xmn, yr = ymx - ymn;
    int gsx, gsy;
    if (xr > yr) {
        gsy = (int)floorf((float)GS * yr / xr);
        if (gsy < 1) gsy = 1;
        gsx = GS;
    } else {
        gsx = (int)floorf((float)GS * xr / yr);
        if (gsx < 1) gsx = 1;
        gsy = GS;
    }
    p.xmin = xmn; p.xmax = xmx; p.ymin = ymn; p.ymax = ymx;
    p.gsx = gsx; p.gsy = gsy;
    return p;
}

// ---------- init: zero the 3 scatter grids, reset minmax/valid for batch b ----------
__global__ void init_kernel(float* __restrict__ grids, unsigned* __restrict__ mm,
                            unsigned* __restrict__ valid, int b) {
    int i = blockIdx.x * blockDim.x + threadIdx.x;
    if (i < 3 * GS2) grids[i] = 0.0f;
    if (i == 0) {
        mm[4 * b + 0] = encf(1e30f);
        mm[4 * b + 1] = encf(-1e30f);
        mm[4 * b + 2] = encf(1e30f);
        mm[4 * b + 3] = encf(-1e30f);
        valid[b] = 0u;
    }
}

// ---------- pass 1: masked min/max of x,y (streaming, block-reduced) ----------
__global__ void minmax_kernel(const float* __restrict__ coords, const float* __restrict__ cbd,
                              int N, unsigned* __restrict__ mm, unsigned* __restrict__ valid, int b) {
    float xmn = 1e30f, xmx = -1e30f, ymn = 1e30f, ymx = -1e30f;
    int any = 0;
    for (int n = blockIdx.x * blockDim.x + threadIdx.x; n < N; n += gridDim.x * blockDim.x) {
        float d = cbd[n];
        if (d < SIGMA_F) {
            any = 1;
            float x = coords[(size_t)n * 3 + 0];
            float y = coords[(size_t)n * 3 + 1];
            xmn = fminf(xmn, x); xmx = fmaxf(xmx, x);
            ymn = fminf(ymn, y); ymx = fmaxf(ymx, y);
        }
    }
    __shared__ float sxm[256], sxM[256], sym[256], syM[256];
    __shared__ int sany[256];
    int t = threadIdx.x;
    sxm[t] = xmn; sxM[t] = xmx; sym[t] = ymn; syM[t] = ymx; sany[t] = any;
    __syncthreads();
    for (int off = 128; off > 0; off >>= 1) {
        if (t < off) {
            sxm[t] = fminf(sxm[t], sxm[t + off]);
            sxM[t] = fmaxf(sxM[t], sxM[t + off]);
            sym[t] = fminf(sym[t], sym[t + off]);
            syM[t] = fmaxf(syM[t], syM[t + off]);
            sany[t] |= sany[t + off];
        }
        __syncthreads();
    }
    if (t == 0) {
        atomicMin(&mm[4 * b + 0], encf(sxm[0]));
        atomicMax(&mm[4 * b + 1], encf(sxM[0]));
        atomicMin(&mm[4 * b + 2], encf(sym[0]));
        atomicMax(&mm[4 * b + 3], encf(syM[0]));
        if (sany[0]) atomicOr(&valid[b], 1u);
    }
}

// ---------- pass 2: argmax over 8 logits + scatter-add into L2-resident grids ----------
__global__ void scatter_kernel(const float* __restrict__ logits, const int* __restrict__ targets,
                               const float* __restrict__ coords, const float* __restrict__ cbd,
                               int N, const unsigned* __restrict__ mm, int b,
                               float* __restrict__ psum, float* __restrict__ tsum,
                               float* __restrict__ cnt) {
    int n = blockIdx.x * blockDim.x + threadIdx.x;
    if (n >= N) return;
    if (!(cbd[n] < SIGMA_F)) return;  // w == 0 contributes nothing; skip atomics
    BP p = get_params(mm, b);
    float x = coords[(size_t)n * 3 + 0];
    float y = coords[(size_t)n * 3 + 1];
    const float4* lg = (const float4*)(logits + (size_t)n * 8);
    float4 l0 = lg[0];
    float4 l1 = lg[1];
    float best = l0.x; int bi = 0;                 // first-max wins, like jnp.argmax
    if (l0.y > best) { best = l0.y; bi = 1; }
    if (l0.z > best) { best = l0.z; bi = 2; }
    if (l0.w > best) { best = l0.w; bi = 3; }
    if (l1.x > best) { best = l1.x; bi = 4; }
    if (l1.y > best) { best = l1.y; bi = 5; }
    if (l1.z > best) { best = l1.z; bi = 6; }
    if (l1.w > best) { best = l1.w; bi = 7; }
    float pred = (float)bi;
    float tgt  = (float)targets[n];
    float fx = (x - p.xmin) / (p.xmax - p.xmin) * (float)(p.gsx - 1);
    float fy = (y - p.ymin) / (p.ymax - p.ymin) * (float)(p.gsy - 1);
    int xi = (int)fx; if (xi < 0) xi = 0; if (xi > p.gsx - 1) xi = p.gsx - 1;
    int yi = (int)fy; if (yi < 0) yi = 0; if (yi > p.gsy - 1) yi = p.gsy - 1;
    int g = yi * GS + xi;
    atomicAdd(&psum[g], pred);
    atomicAdd(&tsum[g], tgt);
    atomicAdd(&cnt[g], 1.0f);
}

// ---------- bilinear resize of gs_y x gs_x sub-grid to 512x512 (mean on the fly) ----------
__global__ void resize_kernel(const float* __restrict__ psum, const float* __restrict__ tsum,
                              const float* __restrict__ cnt, float* __restrict__ pr,
                              float* __restrict__ tr, const unsigned* __restrict__ mm, int b) {
    int i = blockIdx.x * blockDim.x + threadIdx.x;
    if (i >= GS2) return;
    int X = i & (GS - 1);
    int Y = i >> 9;
    BP p = get_params(mm, b);
    float sy = fmaxf(((float)Y + 0.5f) * (float)p.gsy / (float)GS - 0.5f, 0.0f);
    float sx = fmaxf(((float)X + 0.5f) * (float)p.gsx / (float)GS - 0.5f, 0.0f);
    int y0 = (int)floorf(sy);
    int y1 = min(y0 + 1, p.gsy - 1);
    float wy = sy - (float)y0;
    int x0 = (int)floorf(sx);
    int x1 = min(x0 + 1, p.gsx - 1);
    float wx = sx - (float)x0;
    int i00 = y0 * GS + x0, i01 = y0 * GS + x1, i10 = y1 * GS + x0, i11 = y1 * GS + x1;
    float c00 = fmaxf(cnt[i00], 1.0f), c01 = fmaxf(cnt[i01], 1.0f);
    float c10 = fmaxf(cnt[i10], 1.0f), c11 = fmaxf(cnt[i11], 1.0f);
    {
        float g00 = psum[i00] / c00, g01 = psum[i01] / c01;
        float g10 = psum[i10] / c10, g11 = psum[i11] / c11;
        float top = g00 * (1.0f - wx) + g01 * wx;
        float bot = g10 * (1.0f - wx) + g11 * wx;
        pr[i] = top * (1.0f - wy) + bot * wy;
    }
    {
        float g00 = tsum[i00] / c00, g01 = tsum[i01] / c01;
        float g10 = tsum[i10] / c10, g11 = tsum[i11] / c11;
        float top = g00 * (1.0f - wx) + g01 * wx;
        float bot = g10 * (1.0f - wx) + g11 * wx;
        tr[i] = top * (1.0f - wy) + bot * wy;
    }
}

// ---------- Sobel as implicit GEMM on V_WMMA_F32_16X16X4_F32 ----------
// M = 16 output pixels (one row segment), K = 9 taps padded to 12 (3 chunks of K=4),
// N columns 0/1 = SOBEL_X / SOBEL_Y weights (rest zero). Full fp32.
__constant__ float c_SX[9] = {-1.f, 0.f, 1.f, -2.f, 0.f, 2.f, -1.f, 0.f, 1.f};
__constant__ float c_SY[9] = {-1.f, -2.f, -1.f, 0.f, 0.f, 0.f, 1.f, 2.f, 1.f};

__device__ __forceinline__ float tapval(const float* __restrict__ g, int Y, int Xp, int t) {
    if (t >= 9) return 0.0f;
    int dy = t / 3 - 1, dx = t % 3 - 1;
    int yy = Y + dy, xx = Xp + dx;
    return (yy >= 0 && yy < GS && xx >= 0 && xx < GS) ? g[yy * GS + xx] : 0.0f;
}
__device__ __forceinline__ float wgt(int t, int n) {
    if (t >= 9 || n >= 2) return 0.0f;
    return (n == 0) ? c_SX[t] : c_SY[t];
}

__global__ void sobel_kernel(const float* __restrict__ pr, const float* __restrict__ tr,
                             float* __restrict__ partials, int b) {
    (void)b;
    int lane = threadIdx.x & 31;
    int wave = threadIdx.x >> 5;
    int tile = blockIdx.x * 8 + wave;    // 16384 tiles of 16 pixels
    int Y  = tile >> 5;                  // row
    int X0 = (tile & 31) << 4;           // segment start
    int m  = lane & 15;                  // pixel within segment (A-matrix M row)
    int Xp = X0 + m;
    int kh = (lane >> 4) << 1;           // lanes 0-15 -> K{0,1}; lanes 16-31 -> K{2,3}
    int n  = lane & 15;                  // B-matrix column
    v8f accP = {0.f, 0.f, 0.f, 0.f, 0.f, 0.f, 0.f, 0.f};
    v8f accT = {0.f, 0.f, 0.f, 0.f, 0.f, 0.f, 0.f, 0.f};
#pragma unroll
    for (int c = 0; c < 3; ++c) {
        int t0 = 4 * c + kh;
        v2f a_p = { tapval(pr, Y, Xp, t0), tapval(pr, Y, Xp, t0 + 1) };
        v2f a_t = { tapval(tr, Y, Xp, t0), tapval(tr, Y, Xp, t0 + 1) };
        v2f bw  = { wgt(t0, n), wgt(t0 + 1, n) };
        accP = __builtin_amdgcn_wmma_f32_16x16x4_f32(false, a_p, false, bw,
                                                     (short)0, accP, false, false);
        accT = __builtin_amdgcn_wmma_f32_16x16x4_f32(false, a_t, false, bw,
                                                     (short)0, accT, false, false);
    }
    // D layout: VGPR r, lanes 0-15 -> M=r, N=lane; lanes 16-31 -> M=r+8, N=lane-16.
    // gx lives in lanes 0/16 (N=0), gy in lanes 1/17 (N=1).
    float s = 0.0f;
    int src = (lane & 16) + 1;
#pragma unroll
    for (int r = 0; r < 8; ++r) {
        float gxp = accP[r], gxt = accT[r];
        float gyp = __shfl(accP[r], src, 32);
        float gyt = __shfl(accT[r], src, 32);
        float ep = sqrtf(gxp * gxp + gyp * gyp + 1e-8f);
        float et = sqrtf(gxt * gxt + gyt * gyt + 1e-8f);
        if ((lane & 15) == 0) s += fabsf(ep - et);
    }
    for (int off = 16; off > 0; off >>= 1) s += __shfl_down(s, off, 32);
    __shared__ float wsum[8];
    if (lane == 0) wsum[wave] = s;
    __syncthreads();
    if (threadIdx.x == 0) {
        float tot = 0.0f;
        for (int w = 0; w < 8; ++w) tot += wsum[w];
        partials[blockIdx.x] = tot;   // fixed slot -> deterministic
    }
}

// ---------- deterministic per-batch reduction of block partials ----------
__global__ void reduce_partials(const float* __restrict__ partials, float* __restrict__ loss, int b) {
    __shared__ float sm[256];
    float s = 0.0f;
    for (int i = threadIdx.x; i < 2048; i += 256) s += partials[i];
    sm[threadIdx.x] = s;
    __syncthreads();
    for (int off = 128; off > 0; off >>= 1) {
        if (threadIdx.x < off) sm[threadIdx.x] += sm[threadIdx.x + off];
        __syncthreads();
    }
    if (threadIdx.x == 0) loss[b] = sm[0] / (float)GS2;
}

// ---------- validity-weighted combine ----------
__global__ void final_kernel(const float* __restrict__ loss, const unsigned* __restrict__ valid,
                             float* __restrict__ out) {
    if (blockIdx.x == 0 && threadIdx.x == 0) {
        float s = 0.0f, c = 0.0f;
        for (int b = 0; b < 4; ++b) {
            float v = valid[b] ? 1.0f : 0.0f;
            s += loss[b] * v;
            c += v;
        }
        out[0] = (c > 0.0f) ? (s / fmaxf(c, 1.0f)) : 0.0f;
    }
}

extern "C" void kernel_launch(void* const* d_in, const int* in_sizes, int n_in,
                              void* d_out, int out_size, void* d_ws, size_t ws_size,
                              hipStream_t stream) {
    (void)n_in; (void)out_size; (void)ws_size;
    const float* logits  = (const float*)d_in[0];
    const int*   targets = (const int*)d_in[1];
    const float* coords  = (const float*)d_in[2];
    const float* cbd     = (const float*)d_in[3];
    float* out = (float*)d_out;

    const int B = 4;
    const int N = in_sizes[3] / B;

    // workspace layout (float elements); batches processed sequentially so the
    // five 1 MB grids are reused -> ~5.3 MB total scratch.
    float*    wsf      = (float*)d_ws;
    unsigned* mm       = (unsigned*)d_ws;   // [0..15]  4 batches x {xmin,xmax,ymin,ymax} encoded
    unsigned* valid    = mm + 16;           // [16..19]
    float*    loss     = wsf + 20;          // [20..23]
    float*    partials = wsf + 64;          // [64..2111]
    float*    psum     = wsf + 4096;
    float*    tsum     = psum + GS2;
    float*    cnt      = tsum + GS2;
    float*    pr       = cnt + GS2;
    float*    tr       = pr + GS2;

    for (int b = 0; b < B; ++b) {
        const float* lg = logits + (size_t)b * N * 8;
        const int*   tg = targets + (size_t)b * N;
        const float* co = coords + (size_t)b * N * 3;
        const float* cb = cbd + (size_t)b * N;

        init_kernel<<<(3 * GS2 + 255) / 256, 256, 0, stream>>>(psum, mm, valid, b);
        minmax_kernel<<<1024, 256, 0, stream>>>(co, cb, N, mm, valid, b);
        scatter_kernel<<<(N + 255) / 256, 256, 0, stream>>>(lg, tg, co, cb, N, mm, b,
                                                            psum, tsum, cnt);
        resize_kernel<<<(GS2 + 255) / 256, 256, 0, stream>>>(psum, tsum, cnt, pr, tr, mm, b);
        sobel_kernel<<<2048, 256, 0, stream>>>(pr, tr, partials, b);
        reduce_partials<<<1, 256, 0, stream>>>(partials, loss, b);
    }
    final_kernel<<<1, 64, 0, stream>>>(loss, valid, out);
}